// Dis_37434934952363
// MI455X (gfx1250) — compile-verified
//
#include <hip/hip_runtime.h>
#include <math.h>

typedef __attribute__((ext_vector_type(2))) float v2f;
typedef __attribute__((ext_vector_type(4))) float v4f;
typedef __attribute__((ext_vector_type(8))) float v8f;

#define DCOLS  4096                 // columns per row (fixed by the reference)
#define NSEG   4                    // K-split factor (4096 waves total -> fills the GPU)
#define SEGLEN (DCOLS / NSEG)       // 1024 columns per segment
#define TILE_K 64                   // columns staged per tile
#define NTILES (SEGLEN / TILE_K)    // 16 tiles per segment
#define LSTR   68                   // LDS row stride (dwords): 64+4 pad -> conflict-free b64 frag reads

// ---------------------------------------------------------------------------
// Stage 1: one wave per (16-row group, K-segment). Streams its 16x1024 slice
// of x1/x2 through LDS tiles with a register prefetch pipeline and accumulates
// dot/sx/sy via V_WMMA_F32_16X16X4_F32 diagonals. Writes 3 partials per row
// into W (disjoint slots -> deterministic, no atomics).
// ---------------------------------------------------------------------------
__global__ __launch_bounds__(32) void cos_partial_wmma(
    const float* __restrict__ x1, const float* __restrict__ x2,
    float* __restrict__ W, int rows)
{
    __shared__ float l1[16 * LSTR];
    __shared__ float l2[16 * LSTR];

    const int lane = threadIdx.x;                     // 0..31 (wave32)
    const unsigned seg = blockIdx.x & (NSEG - 1);
    const unsigned r0  = (blockIdx.x >> 2) * 16u;     // first row of this group
    const unsigned k0  = seg * SEGLEN;                // first column of this segment

    v8f cdot = {}, csx = {}, csy = {};

    // WMMA f32 16x16x4 fragment addressing: lanes 0..15 -> row=lane, K 0..1;
    // lanes 16..31 -> row=lane-16, K 2..3.
    const int frow  = lane & 15;
    const int foff  = (lane >> 4) << 1;
    const int fbase = frow * LSTR + foff;

    v4f pa[8], pb[8];                                 // prefetch registers (one 16x64 tile pair)

    // prologue: prefetch tile 0 (coalesced: 16 consecutive lanes = one 256B row)
    #pragma unroll
    for (int it = 0; it < 8; ++it) {
        const int idx = it * 32 + lane;
        const int row = idx >> 4, sg = idx & 15;
        const unsigned goff = (r0 + row) * DCOLS + k0 + sg * 4u;
        pa[it] = __builtin_nontemporal_load((const v4f*)(x1 + goff));
        pb[it] = __builtin_nontemporal_load((const v4f*)(x2 + goff));
    }

    for (int t = 0; t < NTILES; ++t) {
        // drain prefetch registers into LDS
        #pragma unroll
        for (int it = 0; it < 8; ++it) {
            const int idx = it * 32 + lane;
            const int row = idx >> 4, sg = idx & 15;
            *(v4f*)(&l1[row * LSTR + sg * 4]) = pa[it];
            *(v4f*)(&l2[row * LSTR + sg * 4]) = pb[it];
        }
        __syncthreads();

        // issue next tile's global loads now: LOADcnt traffic overlaps the
        // DScnt/WMMA compute below (split counters)
        if (t + 1 < NTILES) {
            const unsigned kb = k0 + (unsigned)(t + 1) * TILE_K;
            #pragma unroll
            for (int it = 0; it < 8; ++it) {
                const int idx = it * 32 + lane;
                const int row = idx >> 4, sg = idx & 15;
                const unsigned goff = (r0 + row) * DCOLS + kb + sg * 4u;
                pa[it] = __builtin_nontemporal_load((const v4f*)(x1 + goff));
                pb[it] = __builtin_nontemporal_load((const v4f*)(x2 + goff));
            }
        }

        // 16 k-steps: 2 conflict-free ds_load_b64 + 3 WMMAs each
        #pragma unroll
        for (int kk = 0; kk < TILE_K; kk += 4) {
            v2f a = *(const v2f*)(&l1[fbase + kk]);
            v2f b = *(const v2f*)(&l2[fbase + kk]);
            cdot = __builtin_amdgcn_wmma_f32_16x16x4_f32(false, a, false, b,
                                                         (short)0, cdot, false, false);
            csx  = __builtin_amdgcn_wmma_f32_16x16x4_f32(false, a, false, a,
                                                         (short)0, csx,  false, false);
            csy  = __builtin_amdgcn_wmma_f32_16x16x4_f32(false, b, false, b,
                                                         (short)0, csy,  false, false);
        }
        __syncthreads();
    }

    // diagonal extraction: (i,i) lives at VGPR i, lane i (i<8) and
    // VGPR i-8, lane i+16 (i>=8) -> lanes {0..7, 24..31}, VGPR index lane&7
    const int di = lane & 7;
    float dv = cdot[0], sx = csx[0], sy = csy[0];
    #pragma unroll
    for (int i = 1; i < 8; ++i) {
        if (di == i) { dv = cdot[i]; sx = csx[i]; sy = csy[i]; }
    }
    if ((lane < 8) | (lane >= 24)) {
        const unsigned r = r0 + ((lane < 8) ? (unsigned)lane : (unsigned)(lane - 16));
        W[(0u * NSEG + seg) * (unsigned)rows + r] = dv;
        W[(1u * NSEG + seg) * (unsigned)rows + r] = sx;
        W[(2u * NSEG + seg) * (unsigned)rows + r] = sy;
    }
}

// ---------------------------------------------------------------------------
// Stage 2: deterministic fixed-order segment reduction + cosine epilogue.
// ---------------------------------------------------------------------------
__global__ void cos_finalize(const float* __restrict__ W, float* __restrict__ out, int rows)
{
    const int r = blockIdx.x * blockDim.x + threadIdx.x;
    if (r >= rows) return;
    float dot = 0.f, sx = 0.f, sy = 0.f;
    #pragma unroll
    for (int s = 0; s < NSEG; ++s) {
        dot += W[(0 * NSEG + s) * rows + r];
        sx  += W[(1 * NSEG + s) * rows + r];
        sy  += W[(2 * NSEG + s) * rows + r];
    }
    out[r] = 0.5f * dot / (sqrtf(sx) * sqrtf(sy));
}

// ---------------------------------------------------------------------------
// Fallback (ws too small): round-1 monolithic kernel, one wave per 16 rows.
// ---------------------------------------------------------------------------
__global__ __launch_bounds__(32) void cos_rows_wmma_mono(
    const float* __restrict__ x1, const float* __restrict__ x2,
    float* __restrict__ out)
{
    __shared__ float l1[16 * LSTR];
    __shared__ float l2[16 * LSTR];
    const int lane = threadIdx.x;
    const unsigned r0 = blockIdx.x * 16u;
    v8f cdot = {}, csx = {}, csy = {};
    const int fbase = (lane & 15) * LSTR + ((lane >> 4) << 1);

    for (unsigned kb = 0; kb < DCOLS; kb += TILE_K) {
        #pragma unroll
        for (int it = 0; it < 8; ++it) {
            const int idx = it * 32 + lane;
            const int row = idx >> 4, sg = idx & 15;
            const unsigned goff = (r0 + row) * DCOLS + kb + sg * 4u;
            *(v4f*)(&l1[row * LSTR + sg * 4]) = __builtin_nontemporal_load((const v4f*)(x1 + goff));
            *(v4f*)(&l2[row * LSTR + sg * 4]) = __builtin_nontemporal_load((const v4f*)(x2 + goff));
        }
        __syncthreads();
        #pragma unroll
        for (int kk = 0; kk < TILE_K; kk += 4) {
            v2f a = *(const v2f*)(&l1[fbase + kk]);
            v2f b = *(const v2f*)(&l2[fbase + kk]);
            cdot = __builtin_amdgcn_wmma_f32_16x16x4_f32(false, a, false, b, (short)0, cdot, false, false);
            csx  = __builtin_amdgcn_wmma_f32_16x16x4_f32(false, a, false, a, (short)0, csx,  false, false);
            csy  = __builtin_amdgcn_wmma_f32_16x16x4_f32(false, b, false, b, (short)0, csy,  false, false);
        }
        __syncthreads();
    }
    const int di = lane & 7;
    float dv = cdot[0], sx = csx[0], sy = csy[0];
    #pragma unroll
    for (int i = 1; i < 8; ++i) {
        if (di == i) { dv = cdot[i]; sx = csx[i]; sy = csy[i]; }
    }
    if ((lane < 8) | (lane >= 24)) {
        const unsigned row = (lane < 8) ? (unsigned)lane : (unsigned)(lane - 16);
        out[r0 + row] = 0.5f * dv / (sqrtf(sx) * sqrtf(sy));
    }
}

extern "C" void kernel_launch(void* const* d_in, const int* in_sizes, int n_in,
                              void* d_out, int out_size, void* d_ws, size_t ws_size,
                              hipStream_t stream) {
    const float* x1 = (const float*)d_in[0];
    const float* x2 = (const float*)d_in[1];
    float* out = (float*)d_out;

    const int rows   = in_sizes[0] / DCOLS;     // 16384
    const int groups = rows / 16;               // 1024
    const size_t ws_needed = (size_t)3 * NSEG * (size_t)rows * sizeof(float);  // 768 KB

    if (d_ws != nullptr && ws_size >= ws_needed) {
        float* W = (float*)d_ws;
        hipLaunchKernelGGL(cos_partial_wmma, dim3((unsigned)(groups * NSEG)), dim3(32),
                           0, stream, x1, x2, W, rows);
        hipLaunchKernelGGL(cos_finalize, dim3((unsigned)((rows + 255) / 256)), dim3(256),
                           0, stream, W, out, rows);
    } else {
        hipLaunchKernelGGL(cos_rows_wmma_mono, dim3((unsigned)groups), dim3(32),
                           0, stream, x1, x2, out);
    }
}